// Cache_25391846654085
// MI455X (gfx1250) — compile-verified
//
#include <hip/hip_runtime.h>
#include <math.h>
#include <float.h>

// -------- CDNA5 (gfx1250) wave32 WMMA types --------
typedef float v2f __attribute__((ext_vector_type(2)));
typedef float v4f __attribute__((ext_vector_type(4)));
typedef float v8f __attribute__((ext_vector_type(8)));

#define DIMS 768
#define KTOP 5
#define COS_EPS 1e-8f
#define W_EPS 1e-6f
#define NBLOCKS 1024
#define TPB 256
#define WPB (TPB / 32)   // waves per block (wave32)

// ---------------------------------------------------------------------------
// Kernel 0: query norm -> ws[0]
// ---------------------------------------------------------------------------
__global__ void __launch_bounds__(TPB) qnorm_kernel(const float* __restrict__ q,
                                                    float* __restrict__ qn_out) {
    __shared__ float red[TPB];
    float s = 0.f;
    for (int d = threadIdx.x; d < DIMS; d += TPB) { float v = q[d]; s = fmaf(v, v, s); }
    red[threadIdx.x] = s;
    __syncthreads();
    for (int off = TPB / 2; off > 0; off >>= 1) {
        if (threadIdx.x < off) red[threadIdx.x] += red[threadIdx.x + off];
        __syncthreads();
    }
    if (threadIdx.x == 0) qn_out[0] = fmaxf(sqrtf(red[0]), COS_EPS);
}

// ---------------------------------------------------------------------------
// Kernel 1: streaming cosine distance over 16-row tiles using fp32 WMMA
// (V_WMMA_F32_16X16X4_F32), fused per-lane top-5, block top-5 to candidates.
//
// K-permutation trick: dot(a,b) is invariant to permuting K as long as A and
// B use the same permutation.  We assign WMMA step 2s logical K
// {8s,8s+1 | 8s+4,8s+5} (lower|upper lane half) and step 2s+1
// {8s+2,8s+3 | 8s+6,8s+7}, so each lane feeds TWO WMMA steps from ONE
// contiguous b128 load (global for A, LDS for B).
// ---------------------------------------------------------------------------
__global__ void __launch_bounds__(TPB) simtop_kernel(const float* __restrict__ q,
                                                     const float* __restrict__ db,
                                                     const float* __restrict__ qn_ptr,
                                                     float* __restrict__ candD,
                                                     int* __restrict__ candI,
                                                     int numTiles) {
    __shared__ float qs[DIMS];                 // query staged in LDS (ds_load_b128 feed)
    __shared__ float wD[WPB * 2 * KTOP];       // per-wave-half candidates
    __shared__ int   wI[WPB * 2 * KTOP];

    for (int d = threadIdx.x; d < DIMS; d += TPB) qs[d] = q[d];
    __syncthreads();

    const int lane  = threadIdx.x & 31;
    const int wave  = threadIdx.x >> 5;
    const int gwave = blockIdx.x * WPB + wave;
    const int nwave = gridDim.x * WPB;
    const float qn  = qn_ptr[0];

    const int hlf  = lane >> 4;     // lane half: owns K {..+0..3} vs {..+4..7}
    const int moff = hlf << 3;      // C/D row offset for this half (0 or 8)

    float d5[KTOP]; int i5[KTOP];
#pragma unroll
    for (int j = 0; j < KTOP; ++j) { d5[j] = FLT_MAX; i5[j] = 0; }

    for (int t = gwave; t < numTiles; t += nwave) {
        const int base = t << 4;    // 16 database rows per tile
        const float* rowp = db + (size_t)(base + (lane & 15)) * DIMS + (hlf << 2);
        const float* qp   = qs + (hlf << 2);

        v8f  c = {0.f, 0.f, 0.f, 0.f, 0.f, 0.f, 0.f, 0.f};
        float nsum = 0.f;
#pragma unroll 2
        for (int k = 0; k < DIMS; k += 8) {
            v4f a = *(const v4f*)(rowp + k);   // global_load_b128: 4 contiguous K
            v4f b = *(const v4f*)(qp + k);     // ds_load_b128: matching query K
            nsum = fmaf(a.x, a.x, nsum);
            nsum = fmaf(a.y, a.y, nsum);
            nsum = fmaf(a.z, a.z, nsum);
            nsum = fmaf(a.w, a.w, nsum);
            // two WMMA steps share the one b128 load (same K-permutation on A & B)
            c = __builtin_amdgcn_wmma_f32_16x16x4_f32(
                    false, a.xy, false, b.xy, (short)0, c, false, false);
            c = __builtin_amdgcn_wmma_f32_16x16x4_f32(
                    false, a.zw, false, b.zw, (short)0, c, false, false);
        }

        // full sum-of-squares for row (lane&15): combine the two K-halves
        nsum += __shfl_xor(nsum, 16, 32);
        float dn = fmaxf(sqrtf(nsum), COS_EPS);

#pragma unroll
        for (int r = 0; r < 8; ++r) {
            int   m    = r + moff;                   // row this lane-half owns in C/D
            float dnr  = __shfl(dn, m, 32);          // dn of row m lives in lane m
            float dist = 1.0f - c[r] / (dnr * qn);
            int   idx  = base + m;
            // branch-free sorted insert (keeps EXEC all-1s between WMMAs)
#pragma unroll
            for (int j = KTOP - 1; j >= 0; --j) {
                float dj = d5[j]; int ij = i5[j];
                float dp = (j > 0) ? d5[j - 1] : -FLT_MAX;
                int   ip = (j > 0) ? i5[j - 1] : idx;
                bool lt  = dist < dj;
                bool ltp = dist < dp;
                d5[j] = lt ? (ltp ? dp : dist) : dj;
                i5[j] = lt ? (ltp ? ip : idx)  : ij;
            }
        }
    }

    // lanes 0 and 16 hold the (identical-within-half) top-5 lists
    if ((lane & 15) == 0) {
        int s = (wave * 2 + hlf) * KTOP;
#pragma unroll
        for (int j = 0; j < KTOP; ++j) { wD[s + j] = d5[j]; wI[s + j] = i5[j]; }
    }
    __syncthreads();

    if (threadIdx.x == 0) {   // block-level top-5 over 80 candidates
        const int M = WPB * 2 * KTOP;
        for (int j = 0; j < KTOP; ++j) {
            float best = FLT_MAX; int bi = 0, bp = 0;
            for (int m = 0; m < M; ++m)
                if (wD[m] < best) { best = wD[m]; bi = wI[m]; bp = m; }
            wD[bp] = FLT_MAX;
            candD[blockIdx.x * KTOP + j] = best;
            candI[blockIdx.x * KTOP + j] = bi;
        }
    }
}

// ---------------------------------------------------------------------------
// Kernel 2: global top-5 select, inverse-square-distance weights, centroid.
// ---------------------------------------------------------------------------
__global__ void __launch_bounds__(TPB) final_kernel(const float* __restrict__ db,
                                                    const float* __restrict__ candD,
                                                    const int* __restrict__ candI,
                                                    float* __restrict__ out,
                                                    int nCand) {
    __shared__ float cd[NBLOCKS * KTOP];
    __shared__ int   ci[NBLOCKS * KTOP];
    __shared__ float selW[KTOP];
    __shared__ int   selI[KTOP];

    for (int m = threadIdx.x; m < nCand; m += TPB) { cd[m] = candD[m]; ci[m] = candI[m]; }
    __syncthreads();

    if (threadIdx.x == 0) {
        float dsel[KTOP];
        for (int j = 0; j < KTOP; ++j) {
            float best = FLT_MAX; int bp = 0;
            for (int m = 0; m < nCand; ++m)
                if (cd[m] < best) { best = cd[m]; bp = m; }
            dsel[j] = best; selI[j] = ci[bp]; cd[bp] = FLT_MAX;
        }
        float w[KTOP], wsum = 0.f;
        for (int j = 0; j < KTOP; ++j) {
            float dw = dsel[j] + W_EPS;
            w[j] = 1.0f / (dw * dw);
            wsum += w[j];
        }
        for (int j = 0; j < KTOP; ++j) selW[j] = w[j] / wsum;
    }
    __syncthreads();

    for (int d = threadIdx.x; d < DIMS; d += TPB) {
        float acc = 0.f;
#pragma unroll
        for (int j = 0; j < KTOP; ++j)
            acc = fmaf(selW[j], db[(size_t)selI[j] * DIMS + d], acc);
        out[d] = acc;
    }
}

// ---------------------------------------------------------------------------
extern "C" void kernel_launch(void* const* d_in, const int* in_sizes, int n_in,
                              void* d_out, int out_size, void* d_ws, size_t ws_size,
                              hipStream_t stream) {
    const float* q  = (const float*)d_in[0];   // [1,768] fp32
    const float* db = (const float*)d_in[1];   // [N,768] fp32
    float* out = (float*)d_out;                // [1,768] fp32

    const int N = in_sizes[1] / DIMS;          // 500000 (multiple of 16)
    const int numTiles = N >> 4;

    // ws layout (floats): [0]=qnorm, [64..)=candD (NBLOCKS*KTOP), then candI
    float* ws_f  = (float*)d_ws;
    float* qn    = ws_f;
    float* candD = ws_f + 64;
    int*   candI = (int*)(ws_f + 64 + NBLOCKS * KTOP);

    qnorm_kernel<<<1, TPB, 0, stream>>>(q, qn);
    simtop_kernel<<<NBLOCKS, TPB, 0, stream>>>(q, db, qn, candD, candI, numTiles);
    final_kernel<<<1, TPB, 0, stream>>>(db, candD, candI, out, NBLOCKS * KTOP);
}